// GCN_Net_81432579932420
// MI455X (gfx1250) — compile-verified
//
#include <hip/hip_runtime.h>
#include <math.h>

#define NNODE 50000
#define NEDGE 800000
#define NIN   512
#define NHID  128
#define NCLS  64

typedef __attribute__((ext_vector_type(2))) float v2f;
typedef __attribute__((ext_vector_type(8))) float v8f;

// ---------------- degree / normalization ----------------

__global__ __launch_bounds__(256) void init_deg_kernel(float* __restrict__ deg, int n) {
    int i = blockIdx.x * blockDim.x + threadIdx.x;
    if (i < n) deg[i] = 1.0f;  // +1 self-loop
}

__global__ __launch_bounds__(256) void deg_accum_kernel(const int* __restrict__ ei,
                                                        float* __restrict__ deg, int nE) {
    int e = blockIdx.x * blockDim.x + threadIdx.x;
    if (e < nE) atomicAdd(&deg[ei[nE + e]], 1.0f);  // dst = ei[E + e]
}

__global__ __launch_bounds__(256) void deg_to_dinv_kernel(float* __restrict__ deg, int n) {
    int i = blockIdx.x * blockDim.x + threadIdx.x;
    if (i < n) deg[i] = rsqrtf(deg[i]);  // deg >= 1
}

// ---------------- fp32 WMMA GEMM: out[M x NOUT] = A[M x K] @ W[K x NOUT] ----------------
// One block per 16-row strip; wave w computes the 16x16 tile at columns [16w, 16w+16).
// EXEC is all-1s (M % 16 == 0, full blocks) as required by WMMA.

template <int K, int NOUT>
__global__ __launch_bounds__(256) void gemm_wmma_f32(const float* __restrict__ A,
                                                     const float* __restrict__ W,
                                                     float* __restrict__ out) {
    const int m0   = blockIdx.x << 4;
    const int wave = threadIdx.x >> 5;
    const int lane = threadIdx.x & 31;
    const int half = lane >> 4;   // 0: lanes 0-15, 1: lanes 16-31
    const int mr   = lane & 15;
    const int n0   = wave << 4;

    const float* __restrict__ arow = A + (size_t)(m0 + mr) * K;   // this lane's A row
    const float* __restrict__ wcol = W + n0 + mr;                 // this lane's B column

    v8f c = {};
#pragma unroll 8
    for (int k = 0; k < K; k += 4) {
        // A 16x4 fragment: lanes 0-15 hold K = k..k+1, lanes 16-31 hold K = k+2..k+3
        v2f a;
        a.x = arow[k + 2 * half + 0];
        a.y = arow[k + 2 * half + 1];
        // B 4x16 fragment: same K split across lane halves, N = n0 + (lane & 15)
        const float* wp = wcol + (size_t)(k + 2 * half) * NOUT;
        v2f b;
        b.x = wp[0];
        b.y = wp[NOUT];
        c = __builtin_amdgcn_wmma_f32_16x16x4_f32(false, a, false, b, (short)0, c,
                                                  false, false);
    }

    // D 16x16 f32: VGPR r -> M = 8*half + r, N = n0 + mr
#pragma unroll
    for (int r = 0; r < 8; ++r)
        out[(size_t)(m0 + 8 * half + r) * NOUT + n0 + mr] = c[r];
}

// ---------------- self-loop init (+ optional bias) ----------------

template <int C, bool WITH_BIAS>
__global__ __launch_bounds__(256) void self_init_kernel(const float* __restrict__ feat,
                                                        const float* __restrict__ dinv,
                                                        const float* __restrict__ bias,
                                                        float* __restrict__ acc, int n) {
    long long t = (long long)blockIdx.x * blockDim.x + threadIdx.x;
    if (t >= (long long)n * C) return;
    int i = (int)(t / C), cix = (int)(t % C);
    float di = dinv[i];
    float v = di * di * feat[(size_t)i * C + cix];
    if (WITH_BIAS) v += bias[cix];
    acc[(size_t)i * C + cix] = v;
}

// ---------------- edge scatter: acc[dst] += dinv[s]*dinv[d] * feat[src] ----------------
// Channel-major thread mapping -> C consecutive lanes gather consecutive floats of
// feat[src] (coalesced) and issue coalesced global_atomic_add_f32 to acc[dst].

template <int C>
__global__ __launch_bounds__(256) void edge_scatter_kernel(const float* __restrict__ feat,
                                                           const int* __restrict__ ei,
                                                           const float* __restrict__ dinv,
                                                           float* __restrict__ acc, int nE) {
    long long t = (long long)blockIdx.x * blockDim.x + threadIdx.x;
    if (t >= (long long)nE * C) return;
    int e = (int)(t / C), cix = (int)(t % C);
    int s = ei[e];
    int d = ei[nE + e];
    float w = dinv[s] * dinv[d];
    atomicAdd(&acc[(size_t)d * C + cix], w * feat[(size_t)s * C + cix]);
}

// ---------------- bias + ReLU ----------------

template <int C>
__global__ __launch_bounds__(256) void bias_relu_kernel(float* __restrict__ h,
                                                        const float* __restrict__ bias,
                                                        int n) {
    long long t = (long long)blockIdx.x * blockDim.x + threadIdx.x;
    if (t >= (long long)n * C) return;
    int cix = (int)(t % C);
    float v = h[t] + bias[cix];
    h[t] = v > 0.0f ? v : 0.0f;
}

// ---------------- log-softmax over 64 classes, one wave32 per row, in place ----------------

__global__ __launch_bounds__(256) void logsoftmax64_kernel(float* __restrict__ out, int nrows) {
    int wid  = (int)(((long long)blockIdx.x * blockDim.x + threadIdx.x) >> 5);
    int lane = threadIdx.x & 31;
    if (wid >= nrows) return;
    float* row = out + (size_t)wid * NCLS;
    float v0 = row[lane];
    float v1 = row[lane + 32];
    float m = fmaxf(v0, v1);
#pragma unroll
    for (int off = 16; off > 0; off >>= 1)
        m = fmaxf(m, __shfl_xor(m, off, 32));
    float s = expf(v0 - m) + expf(v1 - m);
#pragma unroll
    for (int off = 16; off > 0; off >>= 1)
        s += __shfl_xor(s, off, 32);
    float lse = m + logf(s);
    row[lane]      = v0 - lse;
    row[lane + 32] = v1 - lse;
}

// ---------------- launch ----------------

extern "C" void kernel_launch(void* const* d_in, const int* in_sizes, int n_in,
                              void* d_out, int out_size, void* d_ws, size_t ws_size,
                              hipStream_t stream) {
    const float* x  = (const float*)d_in[0];   // [N, 512]
    const int*   ei = (const int*)d_in[1];     // [2, E] flat: src = ei[e], dst = ei[E+e]
    const float* W1 = (const float*)d_in[2];   // [512, 128]
    const float* b1 = (const float*)d_in[3];   // [128]
    const float* W2 = (const float*)d_in[4];   // [128, 64]
    const float* b2 = (const float*)d_in[5];   // [64]
    float* out = (float*)d_out;                // [N, 64]

    char* ws = (char*)d_ws;
    float* dinv = (float*)ws;                                   // [N]
    size_t off  = ((size_t)NNODE * sizeof(float) + 255) & ~(size_t)255;
    float* xw = (float*)(ws + off);                             // [N, 128]
    float* h  = (float*)(ws + off + (size_t)NNODE * NHID * 4);  // [N, 128]
    float* hw = xw;  // xw is dead after layer-1 finalize; reuse for h@W2 [N, 64]

    const int T = 256;
    auto blks = [](long long total, int t) { return (unsigned)((total + t - 1) / t); };

    // degrees -> dinv
    init_deg_kernel<<<blks(NNODE, T), T, 0, stream>>>(dinv, NNODE);
    deg_accum_kernel<<<blks(NEDGE, T), T, 0, stream>>>(ei, dinv, NEDGE);
    deg_to_dinv_kernel<<<blks(NNODE, T), T, 0, stream>>>(dinv, NNODE);

    // layer 1: xw = x @ W1  (WMMA f32), aggregate, bias+ReLU
    gemm_wmma_f32<NIN, NHID><<<NNODE / 16, (NHID / 16) * 32, 0, stream>>>(x, W1, xw);
    self_init_kernel<NHID, false><<<blks((long long)NNODE * NHID, T), T, 0, stream>>>(
        xw, dinv, nullptr, h, NNODE);
    edge_scatter_kernel<NHID><<<blks((long long)NEDGE * NHID, T), T, 0, stream>>>(
        xw, ei, dinv, h, NEDGE);
    bias_relu_kernel<NHID><<<blks((long long)NNODE * NHID, T), T, 0, stream>>>(h, b1, NNODE);

    // layer 2: hw = h @ W2 (WMMA f32), aggregate straight into d_out (+b2)
    gemm_wmma_f32<NHID, NCLS><<<NNODE / 16, (NCLS / 16) * 32, 0, stream>>>(h, W2, hw);
    self_init_kernel<NCLS, true><<<blks((long long)NNODE * NCLS, T), T, 0, stream>>>(
        hw, dinv, b2, out, NNODE);
    edge_scatter_kernel<NCLS><<<blks((long long)NEDGE * NCLS, T), T, 0, stream>>>(
        hw, ei, dinv, out, NEDGE);

    // log-softmax per row (one wave32 per row, 2 elems/lane)
    logsoftmax64_kernel<<<blks((long long)NNODE * 32, T), T, 0, stream>>>(out, NNODE);
}